// AttnSCAN_89361089561147
// MI455X (gfx1250) — compile-verified
//
#include <hip/hip_runtime.h>

typedef float v2f __attribute__((ext_vector_type(2)));
typedef float v8f __attribute__((ext_vector_type(8)));

#define NB 128
#define NQ 128
#define NC 1024
#define ND 1024

// slab pitches (floats); both 16B-aligned rows, bank-conflict-free frag reads
#define QP 36    // query slab: [128 q][32 k] padded to 36
#define CP 264   // context slab: [32 k][256 d] padded to 264

// ---------------------------------------------------------------------------
// CDNA5 async global->LDS (tracked by ASYNCcnt), 16B per lane
// ---------------------------------------------------------------------------
__device__ __forceinline__ void async_load_b128(const float* g, unsigned lds_byte)
{
    asm volatile("global_load_async_to_lds_b128 %0, %1, off"
                 :: "v"(lds_byte), "v"(g)
                 : "memory");
}

__device__ __forceinline__ void wait_async()
{
    asm volatile("s_wait_asynccnt 0x0" ::: "memory");
}

// ---------------------------------------------------------------------------
// Kernel 1: attn_z(b,q,c) = 9 * l2norm_over_q( leakyrelu( context @ query^T ) )
// grid (4 c-tiles, 128 b), 256 threads (8 waves). Wave w: 32 c-rows x 128 q
// (16 f32 WMMA accumulators). Query K-slab (128q x 32k) double-buffered into
// LDS via global_load_async_to_lds_b128; context A-frags stream from global.
// ---------------------------------------------------------------------------
__global__ __launch_bounds__(256) void gemm1_lrelu_l2n(
    const float* __restrict__ query,    // (b, q, d)
    const float* __restrict__ context,  // (b, c, d)
    float* __restrict__ attn_z)         // (b, q, c)
{
    extern __shared__ float lds[];      // >= 128*257 floats; slabs overlay front
    const int ctile = blockIdx.x;       // 0..3 (256 c each)
    const int b     = blockIdx.y;
    const int tid   = threadIdx.x;
    const int wave  = tid >> 5;
    const int lane  = tid & 31;
    const int lm    = lane & 15;
    const int lh    = lane >> 4;

    const float* ctx_b = context + (size_t)b * NC * ND;
    const float* q_b   = query   + (size_t)b * NQ * ND;

    const unsigned lds_base = (unsigned)(size_t)(&lds[0]);
    const unsigned slab_off[2] = {0u, (unsigned)(NQ * QP * 4)};

    const int c0 = ctile * 256 + wave * 32;
    // A frags (f32 16x4 layout): context[c0 + m*16 + lm][k + lh*2 + {0,1}]
    const float* aptr0 = ctx_b + (size_t)(c0 + lm) * ND + lh * 2;
    const float* aptr1 = aptr0 + (size_t)16 * ND;

    v8f acc[2][8];
#pragma unroll
    for (int m = 0; m < 2; ++m)
#pragma unroll
        for (int t = 0; t < 8; ++t) acc[m][t] = (v8f)0.0f;

    // ---- preload slab 0: query[0..127][0..31] -> lds slab ----
#pragma unroll
    for (int i = 0; i < 4; ++i) {
        const int linear = i * 256 + tid;       // 0..1023
        const int q  = linear >> 3;
        const int ch = linear & 7;              // 8 x b128 per row
        async_load_b128(q_b + (size_t)q * ND + ch * 4,
                        lds_base + (unsigned)((q * QP + ch * 4) * 4));
    }
    wait_async();
    __syncthreads();

    int buf = 0;
    for (int kk = 0; kk < ND; kk += 32) {
        const bool more = (kk + 32) < ND;
        if (more) {
#pragma unroll
            for (int i = 0; i < 4; ++i) {
                const int linear = i * 256 + tid;
                const int q  = linear >> 3;
                const int ch = linear & 7;
                async_load_b128(q_b + (size_t)q * ND + kk + 32 + ch * 4,
                                slab_off[buf ^ 1] + lds_base +
                                    (unsigned)((q * QP + ch * 4) * 4));
            }
        }
        const float* slab = lds + (buf ? NQ * QP : 0);
        __builtin_prefetch(aptr0 + kk + 32, 0, 0);
        __builtin_prefetch(aptr1 + kk + 32, 0, 0);
#pragma unroll
        for (int k4 = 0; k4 < 32; k4 += 4) {
            v2f a0 = *(const v2f*)(aptr0 + kk + k4);
            v2f a1 = *(const v2f*)(aptr1 + kk + k4);
#pragma unroll
            for (int t = 0; t < 8; ++t) {
                // B frag: query[t*16+lm][kk + k4 + lh*2 + {0,1}] from LDS
                v2f bf = *(const v2f*)(slab + (t * 16 + lm) * QP + k4 + lh * 2);
                acc[0][t] = __builtin_amdgcn_wmma_f32_16x16x4_f32(
                    false, a0, false, bf, (short)0, acc[0][t], false, false);
                acc[1][t] = __builtin_amdgcn_wmma_f32_16x16x4_f32(
                    false, a1, false, bf, (short)0, acc[1][t], false, false);
            }
        }
        if (more) {
            wait_async();
            __syncthreads();
            buf ^= 1;
        }
    }

    // leaky relu (slope 0.1)
#pragma unroll
    for (int m = 0; m < 2; ++m)
#pragma unroll
        for (int t = 0; t < 8; ++t)
#pragma unroll
            for (int j = 0; j < 8; ++j) {
                float x = acc[m][t][j];
                acc[m][t][j] = x >= 0.0f ? x : 0.1f * x;
            }

    // L2 norm over q per c-row (row c = c0 + m*16 + j + lh*8); fold SMOOTH=9
#pragma unroll
    for (int m = 0; m < 2; ++m)
#pragma unroll
        for (int j = 0; j < 8; ++j) {
            float s = 0.0f;
#pragma unroll
            for (int t = 0; t < 8; ++t) s += acc[m][t][j] * acc[m][t][j];
            s += __shfl_xor(s, 1, 32);
            s += __shfl_xor(s, 2, 32);
            s += __shfl_xor(s, 4, 32);
            s += __shfl_xor(s, 8, 32);
            const float scale = 9.0f / (sqrtf(s) + 1e-8f);
#pragma unroll
            for (int t = 0; t < 8; ++t) acc[m][t][j] *= scale;
        }

    // transpose through LDS (reuses slab region; all async/ds reads done)
    __syncthreads();
#pragma unroll
    for (int m = 0; m < 2; ++m)
#pragma unroll
        for (int t = 0; t < 8; ++t)
#pragma unroll
            for (int j = 0; j < 8; ++j) {
                const int cl = wave * 32 + m * 16 + j + lh * 8;   // 0..255
                const int q  = t * 16 + lm;                       // 0..127
                lds[q * 257 + cl] = acc[m][t][j];
            }
    __syncthreads();

    float* out_b = attn_z + (size_t)b * NQ * NC + (size_t)ctile * 256;
    for (int idx = tid; idx < 128 * 256; idx += 256) {
        const int q = idx >> 8;
        const int c = idx & 255;
        out_b[(size_t)q * NC + c] = lds[q * 257 + c];
    }
}

// ---------------------------------------------------------------------------
// Kernel 2: in-place softmax over c per (b,q) row (values already x9).
// one wave per row; 2048 blocks x 256 threads.
// ---------------------------------------------------------------------------
__global__ __launch_bounds__(256) void softmax_rows(float* __restrict__ attn)
{
    const int wave = threadIdx.x >> 5;
    const int lane = threadIdx.x & 31;
    const size_t row = (size_t)blockIdx.x * 8 + wave;   // 0 .. 16383
    float* p = attn + row * NC;

    float v[32];
    float mx = -3.402823466e38f;
#pragma unroll
    for (int i = 0; i < 32; ++i) {
        v[i] = p[i * 32 + lane];
        mx = fmaxf(mx, v[i]);
    }
#pragma unroll
    for (int m = 1; m < 32; m <<= 1) mx = fmaxf(mx, __shfl_xor(mx, m, 32));

    float sum = 0.0f;
#pragma unroll
    for (int i = 0; i < 32; ++i) {
        v[i] = __expf(v[i] - mx);
        sum += v[i];
    }
#pragma unroll
    for (int m = 1; m < 32; m <<= 1) sum += __shfl_xor(sum, m, 32);

    const float inv = 1.0f / sum;
#pragma unroll
    for (int i = 0; i < 32; ++i) p[i * 32 + lane] = v[i] * inv;
}

// ---------------------------------------------------------------------------
// Kernel 3: wcontext(b,q,d) = attn(b,q,c) @ context(b,c,d)
// grid (4 d-tiles, 128 b), 256 threads (8 waves). Wave w: q0=(w&3)*32 (32 q),
// d-half = (w>>2)*128 within the 256-wide d tile. Context K-slab (32c x 256d)
// double-buffered via async-to-LDS; attn A-frags stream from global.
// ---------------------------------------------------------------------------
__global__ __launch_bounds__(256) void gemm2_wcontext(
    const float* __restrict__ attn,     // (b, q, c) softmaxed
    const float* __restrict__ context,  // (b, c, d)
    float* __restrict__ wctx)           // (b, q, d)
{
    extern __shared__ float lds2[];     // 2 * 32*CP floats
    const int dtile = blockIdx.x;       // 0..3 (256 d each)
    const int b     = blockIdx.y;
    const int tid   = threadIdx.x;
    const int wave  = tid >> 5;
    const int lane  = tid & 31;
    const int lm    = lane & 15;
    const int lh    = lane >> 4;

    const float* ctx_b  = context + (size_t)b * NC * ND;
    const float* attn_b = attn    + (size_t)b * NQ * NC;

    const unsigned lds_base = (unsigned)(size_t)(&lds2[0]);
    const unsigned slab_bytes = (unsigned)(32 * CP * 4);

    const int q0 = (wave & 3) * 32;
    const int d0 = dtile * 256;
    const int dd = (wave >> 2) * 128;

    // A frags: attn[q0 + m*16 + lm][k + lh*2 + {0,1}]
    const float* aptr0 = attn_b + (size_t)(q0 + lm) * NC + lh * 2;
    const float* aptr1 = aptr0 + (size_t)16 * NC;

    v8f acc[2][8];
#pragma unroll
    for (int m = 0; m < 2; ++m)
#pragma unroll
        for (int t = 0; t < 8; ++t) acc[m][t] = (v8f)0.0f;

    // ---- preload slab 0: context[0..31][d0 .. d0+255] ----
#pragma unroll
    for (int i = 0; i < 8; ++i) {
        const int linear = i * 256 + tid;       // 0..2047
        const int k  = linear >> 6;
        const int ch = linear & 63;             // 64 x b128 per row
        async_load_b128(ctx_b + (size_t)k * ND + d0 + ch * 4,
                        lds_base + (unsigned)((k * CP + ch * 4) * 4));
    }
    wait_async();
    __syncthreads();

    int buf = 0;
    for (int kk = 0; kk < NC; kk += 32) {
        const bool more = (kk + 32) < NC;
        if (more) {
#pragma unroll
            for (int i = 0; i < 8; ++i) {
                const int linear = i * 256 + tid;
                const int k  = linear >> 6;
                const int ch = linear & 63;
                async_load_b128(ctx_b + (size_t)(kk + 32 + k) * ND + d0 + ch * 4,
                                lds_base + (buf ? 0u : slab_bytes) +
                                    (unsigned)((k * CP + ch * 4) * 4));
            }
        }
        const float* slab = lds2 + (buf ? 32 * CP : 0);
        __builtin_prefetch(aptr0 + kk + 32, 0, 0);
        __builtin_prefetch(aptr1 + kk + 32, 0, 0);
#pragma unroll
        for (int k4 = 0; k4 < 32; k4 += 4) {
            v2f a0 = *(const v2f*)(aptr0 + kk + k4);
            v2f a1 = *(const v2f*)(aptr1 + kk + k4);
            const float* srow = slab + (k4 + lh * 2) * CP + dd + lm;
#pragma unroll
            for (int t = 0; t < 8; ++t) {
                // B frag: context[kk+k4+lh*2 + {0,1}][d0+dd+t*16+lm] from LDS
                v2f bf;
                bf.x = srow[t * 16];
                bf.y = srow[t * 16 + CP];
                acc[0][t] = __builtin_amdgcn_wmma_f32_16x16x4_f32(
                    false, a0, false, bf, (short)0, acc[0][t], false, false);
                acc[1][t] = __builtin_amdgcn_wmma_f32_16x16x4_f32(
                    false, a1, false, bf, (short)0, acc[1][t], false, false);
            }
        }
        if (more) {
            wait_async();
            __syncthreads();
            buf ^= 1;
        }
    }

    float* out_b = wctx + (size_t)b * NQ * ND + d0 + dd;
#pragma unroll
    for (int m = 0; m < 2; ++m)
#pragma unroll
        for (int t = 0; t < 8; ++t)
#pragma unroll
            for (int j = 0; j < 8; ++j) {
                const int q = q0 + m * 16 + j + lh * 8;
                const int d = t * 16 + lm;
                out_b[(size_t)q * ND + d] = acc[m][t][j];
            }
}

// ---------------------------------------------------------------------------
extern "C" void kernel_launch(void* const* d_in, const int* in_sizes, int n_in,
                              void* d_out, int out_size, void* d_ws, size_t ws_size,
                              hipStream_t stream)
{
    const float* query   = (const float*)d_in[0];  // (128,128,1024)
    const float* context = (const float*)d_in[1];  // (128,1024,1024)

    float* out = (float*)d_out;
    const size_t QSZ = (size_t)NB * NQ * ND;  // 16,777,216 elements each
    float* out_query = out;
    float* out_wctx  = out + QSZ;
    float* out_attn  = out + 2 * QSZ;

    // output 0: passthrough query
    hipMemcpyAsync(out_query, query, QSZ * sizeof(float),
                   hipMemcpyDeviceToDevice, stream);

    // output 2 (staging): z = 9 * l2norm(leakyrelu(context @ query^T)) in (b,q,c)
    gemm1_lrelu_l2n<<<dim3(4, NB), 256, 128 * 257 * sizeof(float), stream>>>(
        query, context, out_attn);

    // output 2 (final): softmax over c, in place
    softmax_rows<<<dim3(NB * NQ / 8), 256, 0, stream>>>(out_attn);

    // output 1: wcontext = attn @ context
    gemm2_wcontext<<<dim3(4, NB), 256, 2 * 32 * CP * sizeof(float), stream>>>(
        out_attn, context, out_wctx);
}